// TDconvD_72533407695194
// MI455X (gfx1250) — compile-verified
//
#include <hip/hip_runtime.h>

// ---------------------------------------------------------------------------
// TDconvD decoder forward, MI455X (gfx1250, wave32, WMMA + async-to-LDS,
// double-buffered).
//
// Math notes (exact-in-real-arithmetic refactorings):
//  * softmax over T of (fscore[b,t] + dscore[b,s] + bp) == softmax(fscore[b,:])
//    (s-terms constant over the reduction axis) -> attention weights are
//    s-independent; Wdec/bdec/bp/benc cancel and ctx is per-b (rank-1 in s).
//  * logits = ctx@W2[:, :E]^T (precomputed per-b "cpart"+b2)
//           + decode@W2[:, E:]^T  (dominant GEMM: 1984x30000x512, mem-bound).
//  * Convs are GEMMs over overlapping rows of a left-zero-padded [B][L+6][D]
//    activation slab against weights repacked to [O][k*512+i].
// fp32 WMMA (v_wmma_f32_16x16x4_f32) keeps reference precision; the dominant
// kernel is HBM-bound (~300MB @ 23.3TB/s ~ 13us) so fp32 matrix rate is ample.
// Tile staging uses gfx1250 GLOBAL_LOAD_ASYNC_TO_LDS_B128 (ASYNCcnt) with two
// LDS buffers so the next tile's DMA overlaps the current WMMA loop.
// ---------------------------------------------------------------------------

typedef __attribute__((ext_vector_type(2))) float v2f;
typedef __attribute__((ext_vector_type(8))) float v8f;

#define BM 64
#define BN 128
#define BK 32
#define BKP 36   // padded LDS row (floats): 144B rows -> 16B aligned, bank-spread
#define ATILE (BM * BKP)
#define BTILE (BN * BKP)

// one lane copies 16 bytes global -> LDS, tracked by ASYNCcnt
__device__ __forceinline__ void async_copy16(unsigned lds_addr, const float* g) {
    asm volatile("global_load_async_to_lds_b128 %0, %1, off"
                 :: "v"(lds_addr), "v"((unsigned long long)(size_t)g)
                 : "memory");
}
__device__ __forceinline__ void wait_async0() {
    asm volatile("s_wait_asynccnt 0x0" ::: "memory");
}

// C[m, n] = sum_k Arow(m)[k] * Bm[n*ldb + k]  (+ bias[n]) (+ add[m/rpb, n])
// Arow(m) = A + (m/rpb)*a_bs + (m%rpb)*a_rs ; C row likewise with c_bs/c_rs.
// Requires: M % 64 == 0 (grid.y = M/64); K % 32 == 0. N arbitrary (guarded).
__global__ __launch_bounds__(256) void gemm_wmma_f32(
    const float* __restrict__ A, int rpb, long a_bs, long a_rs,
    const float* __restrict__ Bm, long ldb,
    const float* __restrict__ bias,
    const float* __restrict__ add, long add_bs,
    float* __restrict__ C, long c_bs, long c_rs,
    int Kdim, int N)
{
    __shared__ __align__(16) float As[2 * ATILE];
    __shared__ __align__(16) float Bs[2 * BTILE];

    const int tid  = threadIdx.x;
    const int wave = tid >> 5;
    const int lane = tid & 31;
    const int m_blk = blockIdx.y * BM;
    const int n_blk = blockIdx.x * BN;
    const int wm = (wave & 1) * 32;        // wave tile: 32(M) x 32(N)
    const int wn = (wave >> 1) * 32;

    // cooperative staging: 256 lanes, 16B each; A: 64x8 chunks (2/lane),
    // B: 128x8 chunks (4/lane)
    const int lrow = tid >> 3;             // 0..31
    const int lcol = (tid & 7) << 2;       // float column 0,4,...,28

    const int am0 = m_blk + lrow, am1 = m_blk + lrow + 32;
    const float* arp0 = A + (long)(am0 / rpb) * a_bs + (long)(am0 % rpb) * a_rs + lcol;
    const float* arp1 = A + (long)(am1 / rpb) * a_bs + (long)(am1 % rpb) * a_rs + lcol;

    const float* brp[4];
#pragma unroll
    for (int j = 0; j < 4; ++j) {
        int bn = n_blk + lrow + j * 32;
        if (bn > N - 1) bn = N - 1;        // clamp: garbage cols masked at store
        brp[j] = Bm + (long)bn * ldb + lcol;
    }

    const unsigned aL0 = (unsigned)(size_t)(As + lrow * BKP + lcol);
    const unsigned aL1 = (unsigned)(size_t)(As + (lrow + 32) * BKP + lcol);
    unsigned bL[4];
#pragma unroll
    for (int j = 0; j < 4; ++j)
        bL[j] = (unsigned)(size_t)(Bs + (lrow + j * 32) * BKP + lcol);

    // issue one K-chunk's tile DMA into LDS buffer `buf`
    auto issue = [&](int buf, int kt) {
        const unsigned ao = buf * (ATILE * 4u);   // byte offsets within LDS
        const unsigned bo = buf * (BTILE * 4u);
        async_copy16(aL0 + ao, arp0 + kt);
        async_copy16(aL1 + ao, arp1 + kt);
        async_copy16(bL[0] + bo, brp[0] + kt);
        async_copy16(bL[1] + bo, brp[1] + kt);
        async_copy16(bL[2] + bo, brp[2] + kt);
        async_copy16(bL[3] + bo, brp[3] + kt);
    };

    // WMMA fragment lane mapping (ISA 7.12.2, 32-bit A 16x4 / B 4x16):
    // lane = (k>>1)*16 + (m|n), vgpr = k&1  -> per-lane float2 at k base.
    const int lm = lane & 15;
    const int lk = (lane >> 4) << 1;

    v8f acc00 = {0.f,0.f,0.f,0.f,0.f,0.f,0.f,0.f};
    v8f acc01 = acc00, acc10 = acc00, acc11 = acc00;

    const int nchunks = Kdim / BK;
    issue(0, 0);                            // prologue: prime buffer 0
    for (int i = 0; i < nchunks; ++i) {
        const int cur = i & 1;
        wait_async0();                      // own DMA into buf[cur] landed
        __syncthreads();                    // all waves' DMA landed; all waves
                                            // done reading buf[cur^1]
        if (i + 1 < nchunks) issue(cur ^ 1, (i + 1) * BK);  // overlaps compute

        const float* Ab0 = As + cur * ATILE + (wm + lm) * BKP + lk;
        const float* Ab1 = Ab0 + 16 * BKP;
        const float* Bb0 = Bs + cur * BTILE + (wn + lm) * BKP + lk;
        const float* Bb1 = Bb0 + 16 * BKP;
#pragma unroll
        for (int kk = 0; kk < BK; kk += 4) {
            v2f a0 = *(const v2f*)(Ab0 + kk);
            v2f a1 = *(const v2f*)(Ab1 + kk);
            v2f b0 = *(const v2f*)(Bb0 + kk);
            v2f b1 = *(const v2f*)(Bb1 + kk);
            acc00 = __builtin_amdgcn_wmma_f32_16x16x4_f32(false, a0, false, b0, (short)0, acc00, false, false);
            acc01 = __builtin_amdgcn_wmma_f32_16x16x4_f32(false, a0, false, b1, (short)0, acc01, false, false);
            acc10 = __builtin_amdgcn_wmma_f32_16x16x4_f32(false, a1, false, b0, (short)0, acc10, false, false);
            acc11 = __builtin_amdgcn_wmma_f32_16x16x4_f32(false, a1, false, b1, (short)0, acc11, false, false);
        }
    }

    // C/D layout: VGPR r -> M = r + 8*(lane>>4); N = lane&15.
    const v8f accs[2][2] = {{acc00, acc01}, {acc10, acc11}};
#pragma unroll
    for (int j = 0; j < 2; ++j) {
        const int n_g = n_blk + wn + j * 16 + lm;
        if (n_g >= N) continue;
        const float bval = bias ? bias[n_g] : 0.0f;
#pragma unroll
        for (int i = 0; i < 2; ++i) {
#pragma unroll
            for (int r = 0; r < 8; ++r) {
                const int m_g = m_blk + wm + i * 16 + r + ((lane >> 4) << 3);
                const long bb = m_g / rpb;
                const long rr = m_g % rpb;
                float v = accs[i][j][r] + bval;
                if (add) v += add[bb * add_bs + n_g];
                C[bb * c_bs + rr * c_rs + n_g] = v;
            }
        }
    }
}

// ----------------------------- small kernels -------------------------------

// enc_mean[b][e] = mean_t features[b][t][e]
__global__ void k_mean(const float* __restrict__ f, float* __restrict__ out) {
    int i = blockIdx.x * 256 + threadIdx.x;          // 64*1024
    int b = i >> 10, e = i & 1023;
    const float* p = f + (long)b * 65536 + e;
    float s = 0.f;
#pragma unroll 8
    for (int t = 0; t < 64; ++t) s += p[t * 1024];
    out[i] = s * (1.0f / 64.0f);
}

// x[(b,l)][0:1024]=enc_mean[b]; x[(b,l)][1024:1536]=embed_W[captions[b,l]]
__global__ void k_build_x(const float* __restrict__ enc_mean,
                          const int* __restrict__ captions,
                          const float* __restrict__ embed_W,
                          float* __restrict__ x) {
    int i = blockIdx.x * 256 + threadIdx.x;          // 2048*1536
    int col = i % 1536;
    int row = i / 1536;
    int b = row >> 5, l = row & 31;
    float v;
    if (col < 1024) {
        v = enc_mean[b * 1024 + col];
    } else {
        int tok = captions[b * 32 + l];
        v = embed_W[(long)tok * 512 + (col - 1024)];
    }
    x[i] = v;
}

// zero the 6 left-pad rows of each batch's [38][512] activation slab
__global__ void k_zero_pad(float* __restrict__ xpad) {
    int i = blockIdx.x * 256 + threadIdx.x;          // 64*6*512
    int b = i / 3072, r = i % 3072;
    xpad[(long)b * 19456 + r] = 0.f;
}

// Wc[o][kk*512+c] = conv_w[o][c][kk]   (conv_w: [1024][512][7])
__global__ void k_pack_convw(const float* __restrict__ w, float* __restrict__ Wc) {
    int i = blockIdx.x * 256 + threadIdx.x;          // 1024*3584
    int o = i / 3584;
    int rem = i - o * 3584;
    int kk = rem >> 9, c = rem & 511;
    Wc[i] = w[(long)o * 3584 + c * 7 + kk];
}

// GLU + residual: out[b][l][d] = y[b,l,d]*sigmoid(y[b,l,512+d]) + res[b][l][d]
__global__ void k_glu(const float* __restrict__ y,
                      const float* __restrict__ res,   // base at body row l=0
                      float* __restrict__ out, long out_bs) {
    int i = blockIdx.x * 256 + threadIdx.x;          // 64*32*512
    int b = i >> 14;
    int rem = i & 16383;
    int l = rem >> 9, d = rem & 511;
    const float* yr = y + (long)(b * 32 + l) * 1024;
    float g = yr[d], z = yr[512 + d];
    float r = res[(long)b * 19456 + l * 512 + d];
    out[(long)b * out_bs + l * 512 + d] = g / (1.0f + __expf(-z)) + r;
}

// wfuse[e] = sum_a Wenc[a][e] * Wp[a]
__global__ void k_wfuse(const float* __restrict__ Wenc,
                        const float* __restrict__ Wp,
                        float* __restrict__ wfuse) {
    int e = blockIdx.x * 256 + threadIdx.x;          // 1024
    float s = 0.f;
    for (int a = 0; a < 512; ++a) s += Wenc[(long)a * 1024 + e] * Wp[a];
    wfuse[e] = s;
}

// fscore[(b,t)] = features[(b,t)] . wfuse
__global__ __launch_bounds__(128) void k_fscore(const float* __restrict__ feat,
                                                const float* __restrict__ wfuse,
                                                float* __restrict__ fscore) {
    __shared__ float sm[128];
    int row = blockIdx.x;                            // 4096
    const float* f = feat + (long)row * 1024;
    float s = 0.f;
    for (int e = threadIdx.x; e < 1024; e += 128) s += f[e] * wfuse[e];
    sm[threadIdx.x] = s;
    __syncthreads();
    for (int st = 64; st > 0; st >>= 1) {
        if (threadIdx.x < st) sm[threadIdx.x] += sm[threadIdx.x + st];
        __syncthreads();
    }
    if (threadIdx.x == 0) fscore[row] = sm[0];
}

// per-b softmax over T=64
__global__ __launch_bounds__(64) void k_softmax(const float* __restrict__ fscore,
                                                float* __restrict__ attw) {
    __shared__ float sm[64];
    int b = blockIdx.x, t = threadIdx.x;
    float v = fscore[b * 64 + t];
    sm[t] = v;
    __syncthreads();
    for (int s = 32; s > 0; s >>= 1) {
        if (t < s) sm[t] = fmaxf(sm[t], sm[t + s]);
        __syncthreads();
    }
    float mx = sm[0];
    __syncthreads();
    float e = __expf(v - mx);
    sm[t] = e;
    __syncthreads();
    for (int s = 32; s > 0; s >>= 1) {
        if (t < s) sm[t] += sm[t + s];
        __syncthreads();
    }
    attw[b * 64 + t] = e / sm[0];
}

// ctx[b][e] = sum_t attw[b][t] * features[b][t][e]
__global__ void k_ctx(const float* __restrict__ feat,
                      const float* __restrict__ attw,
                      float* __restrict__ ctx) {
    int i = blockIdx.x * 256 + threadIdx.x;          // 64*1024
    int b = i >> 10, e = i & 1023;
    const float* f = feat + (long)b * 65536 + e;
    const float* w = attw + b * 64;
    float s = 0.f;
#pragma unroll 8
    for (int t = 0; t < 64; ++t) s += w[t] * f[t * 1024];
    ctx[i] = s;
}

// ------------------------------- launcher ----------------------------------

extern "C" void kernel_launch(void* const* d_in, const int* in_sizes, int n_in,
                              void* d_out, int out_size, void* d_ws, size_t ws_size,
                              hipStream_t stream) {
    const float* features = (const float*)d_in[0];   // [64,64,1024]
    const int*   captions = (const int*)  d_in[1];   // [64,32]
    // d_in[2] lengths: unused by the reference's math
    const float* embed_W  = (const float*)d_in[3];   // [30000,512]
    const float* W1       = (const float*)d_in[4];   // [512,1536]
    const float* b1       = (const float*)d_in[5];   // [512]
    const float* conv_w1  = (const float*)d_in[6];   // [1024,512,7]
    const float* conv_b1  = (const float*)d_in[7];   // [1024]
    const float* conv_w2  = (const float*)d_in[8];
    const float* conv_b2  = (const float*)d_in[9];
    const float* Wenc     = (const float*)d_in[10];  // [512,1024]
    // benc/Wdec/bdec/bp cancel inside softmax (shift-invariance); unused.
    const float* Wp       = (const float*)d_in[14];  // [512]
    const float* W2       = (const float*)d_in[16];  // [30000,1536]
    const float* b2       = (const float*)d_in[17];  // [30000]
    float* out = (float*)d_out;                      // [64,31,30000]
    float* ws  = (float*)d_ws;

    // workspace layout (floats)
    float* enc_mean = ws;                    // 65536
    float* xbuf     = ws + 65536;            // 3145728 (dead after GEMM1)
    float* ybuf     = xbuf;                  // 2097152, aliases xbuf
    float* xpad1    = ws + 3211264;          // 64*38*512 = 1245184
    float* xpad2    = ws + 4456448;          // 1245184
    float* x3       = ws + 5701632;          // 64*32*512 = 1048576
    float* Wc       = ws + 6750208;          // 1024*3584 = 3670016
    float* fscore   = ws + 10420224;         // 4096
    float* attw     = ws + 10424320;         // 4096
    float* ctxb     = ws + 10428416;         // 65536
    float* cpart    = ws + 10493952;         // 64*30000 = 1920000
    float* wfuse    = ws + 12413952;         // 1024
    (void)ws_size; (void)n_in; (void)in_sizes; (void)out_size;

    // 1) enc_mean
    k_mean<<<256, 256, 0, stream>>>(features, enc_mean);
    // 2) x = concat(enc_mean, embed)
    k_build_x<<<12288, 256, 0, stream>>>(enc_mean, captions, embed_W, xbuf);
    // 3) zero pad rows of both conv input slabs
    k_zero_pad<<<768, 256, 0, stream>>>(xpad1);
    k_zero_pad<<<768, 256, 0, stream>>>(xpad2);
    // 4) GEMM1: h0 = x @ W1^T + b1  -> xpad1 body rows (offset 6 rows per b)
    gemm_wmma_f32<<<dim3(512 / BN, 2048 / BM), 256, 0, stream>>>(
        xbuf, 32, 32L * 1536, 1536,
        W1, 1536, b1, nullptr, 0,
        xpad1 + 6 * 512, 38L * 512, 512,
        1536, 512);
    // 5) conv1 as GEMM over sliding windows + GLU + residual
    k_pack_convw<<<14336, 256, 0, stream>>>(conv_w1, Wc);
    gemm_wmma_f32<<<dim3(1024 / BN, 2048 / BM), 256, 0, stream>>>(
        xpad1, 32, 38L * 512, 512,
        Wc, 3584, conv_b1, nullptr, 0,
        ybuf, 32L * 1024, 1024,
        3584, 1024);
    k_glu<<<4096, 256, 0, stream>>>(ybuf, xpad1 + 6 * 512, xpad2 + 6 * 512, 19456);
    // 6) conv2 + GLU -> x3 (= decode rows, [B][32][512], use l=0..30)
    k_pack_convw<<<14336, 256, 0, stream>>>(conv_w2, Wc);
    gemm_wmma_f32<<<dim3(1024 / BN, 2048 / BM), 256, 0, stream>>>(
        xpad2, 32, 38L * 512, 512,
        Wc, 3584, conv_b2, nullptr, 0,
        ybuf, 32L * 1024, 1024,
        3584, 1024);
    k_glu<<<4096, 256, 0, stream>>>(ybuf, xpad2 + 6 * 512, x3, 16384);
    // 7) attention (rank-1 in s): fscore -> softmax over T -> ctx[b]
    k_wfuse<<<4, 256, 0, stream>>>(Wenc, Wp, wfuse);
    k_fscore<<<4096, 128, 0, stream>>>(features, wfuse, fscore);
    k_softmax<<<64, 64, 0, stream>>>(fscore, attw);
    k_ctx<<<256, 256, 0, stream>>>(features, attw, ctxb);
    // 8) cpart[b][v] = ctx[b] @ W2[:, :1024]^T + b2
    gemm_wmma_f32<<<dim3((30000 + BN - 1) / BN, 64 / BM), 256, 0, stream>>>(
        ctxb, 64, 64L * 1024, 1024,
        W2, 1536, b2, nullptr, 0,
        cpart, 64L * 30000, 30000,
        1024, 30000);
    // 9) logits[(b,s)][v] = decode[(b,s)] @ W2[:, 1024:]^T + cpart[b][v]
    gemm_wmma_f32<<<dim3((30000 + BN - 1) / BN, 1984 / BM), 256, 0, stream>>>(
        x3, 31, 32L * 512, 512,
        W2 + 1024, 1536, nullptr, cpart, 30000,
        out, 31L * 30000, 30000,
        512, 30000);
}